// NGCF_223338299967
// MI455X (gfx1250) — compile-verified
//
#include <hip/hip_runtime.h>
#include <hip/hip_bf16.h>

// NGCF forward for MI455X (gfx1250, wave32).
// - SpMM scatter is the dominant cost; working set (~100 MB) is L2-resident
//   (192 MB L2), atomics are f32 adds at minimum granularity.
// - Dense 64x64 GEMMs run on V_WMMA_F32_16X16X4_F32 (full f32 fidelity).
//   Each wave produces a full 16x64 row-tile (4 accumulators, 128 WMMAs),
//   A fragments gathered once and reused 4x. Weights are staged in LDS in a
//   K-pair-interleaved layout so each B fragment is one ds_load_b64 (no
//   register repacking in the inner loop).

#define USER_NUM 60000
#define ITEM_NUM 40000
#define NODE_NUM (USER_NUM + ITEM_NUM)
#define EMBED_DIM 64
#define N_LAYER 3
#define NNZ 1600000
#define BATCH 4096
#define NEG_SLOPE 0.2f
#define NODE_TILES (NODE_NUM / 16)   // 6250, exact

typedef float v2f __attribute__((ext_vector_type(2)));
typedef float v8f __attribute__((ext_vector_type(8)));

// ---------------------------------------------------------------- zero side_E
__global__ void ngcf_zero_kernel(float4* __restrict__ p, int n4) {
    int i = blockIdx.x * blockDim.x + threadIdx.x;
    if (i < n4) p[i] = make_float4(0.f, 0.f, 0.f, 0.f);
}

// ---------------------------------------------------------------- SpMM (COO)
// One wave per nonzero. Edge index is wave-uniform -> val/row/col become
// scalar (SMEM) loads. Each lane handles 2 of the 64 dims: float2 gather from
// E[col], scale, and two global_atomic_add_f32 into side[row].
__global__ void ngcf_spmm_kernel(const float* __restrict__ vals,
                                 const int*   __restrict__ rows,
                                 const int*   __restrict__ cols,
                                 const float* __restrict__ E,
                                 float*       __restrict__ side) {
    const int wavesPerBlock = blockDim.x >> 5;
    const int edge = blockIdx.x * wavesPerBlock + (threadIdx.x >> 5);
    if (edge >= NNZ) return;
    const int lane = threadIdx.x & 31;

    const float v = vals[edge];
    const int   r = rows[edge];
    const int   c = cols[edge];

    const float2 e = *(const float2*)(E + (size_t)c * EMBED_DIM + lane * 2);
    float* dst = side + (size_t)r * EMBED_DIM + lane * 2;
    atomicAdd(dst + 0, v * e.x);
    atomicAdd(dst + 1, v * e.y);
}

// ------------------------------------------------- fused layer (WMMA GEMMs)
// Per wave: one 16x64 tile of E_next (4 x v8f accumulators, 128 WMMAs).
//   sum_E = side + prev ; bi_E = prev * side   (built in-register in A layout)
//   acc[nb] += sum_E(16x4) @ W1(4x16_nb)  +  bi_E(16x4) @ W2(4x16_nb)
//   E_next = leaky_relu(acc)
// f32 16x4 A layout: lanes 0-15 carry (M=lane, K=k0,k0+1); lanes 16-31 carry
// (M=lane-16, K=k0+2,k0+3). B 4x16 mirrored: lanes give N, vgpr pair gives K.
// LDS weight image is K-pair interleaved: wi[(k>>1)*128 + n*2 + (k&1)], so a
// lane's B fragment {W[kb][n], W[kb+1][n]} (kb even) is one aligned float2.
__global__ void ngcf_layer_kernel(const float* __restrict__ Eprev,
                                  const float* __restrict__ side,
                                  const float* __restrict__ W1,   // 64x64 (this layer)
                                  const float* __restrict__ W2,   // 64x64 (this layer)
                                  float*       __restrict__ Eout) {
    __shared__ float w1i[EMBED_DIM * EMBED_DIM];
    __shared__ float w2i[EMBED_DIM * EMBED_DIM];
    for (int i = threadIdx.x; i < EMBED_DIM * EMBED_DIM; i += blockDim.x) {
        const int k = i >> 6;
        const int n = i & 63;
        const int dst = ((k >> 1) << 7) + (n << 1) + (k & 1);
        w1i[dst] = W1[i];
        w2i[dst] = W2[i];
    }
    __syncthreads();

    const int tile = blockIdx.x * (blockDim.x >> 5) + (threadIdx.x >> 5);
    if (tile >= NODE_TILES) return;            // wave-uniform: EXEC stays all-1s

    const int lane    = threadIdx.x & 31;
    const int halfsel = lane >> 4;             // 0: lanes 0-15, 1: lanes 16-31
    const int nlane   = lane & 15;
    const int m0      = tile * 16;
    const int mrow    = m0 + nlane;
    const int koff    = halfsel * 2;

    v8f acc[4] = {v8f{}, v8f{}, v8f{}, v8f{}};

    const float* epRow = Eprev + (size_t)mrow * EMBED_DIM + koff;
    const float* sdRow = side  + (size_t)mrow * EMBED_DIM + koff;

#pragma unroll
    for (int kk = 0; kk < 16; ++kk) {
        const int k0 = kk * 4;
        const float2 ep = *(const float2*)(epRow + k0);
        const float2 sd = *(const float2*)(sdRow + k0);

        v2f aSum, aBi;
        aSum.x = ep.x + sd.x;  aSum.y = ep.y + sd.y;   // sum_E fragment
        aBi.x  = ep.x * sd.x;  aBi.y  = ep.y * sd.y;   // bi_E fragment

        // kb = k0 + koff (even); K-pair row index = kb/2 = k0/2 + halfsel.
        const int kph = ((k0 >> 1) + halfsel) << 7;    // *128 floats per pair-row
        const float* w1p = w1i + kph + (nlane << 1);
        const float* w2p = w2i + kph + (nlane << 1);

#pragma unroll
        for (int nb = 0; nb < 4; ++nb) {
            const v2f b1 = *(const v2f*)(w1p + nb * 32);
            const v2f b2 = *(const v2f*)(w2p + nb * 32);
            acc[nb] = __builtin_amdgcn_wmma_f32_16x16x4_f32(
                false, aSum, false, b1, (short)0, acc[nb], false, false);
            acc[nb] = __builtin_amdgcn_wmma_f32_16x16x4_f32(
                false, aBi,  false, b2, (short)0, acc[nb], false, false);
        }
    }

    // C/D layout: vgpr i, lane -> row m0 + i + halfsel*8, col nb*16 + nlane.
#pragma unroll
    for (int nb = 0; nb < 4; ++nb) {
#pragma unroll
        for (int i = 0; i < 8; ++i) {
            float v = acc[nb][i];
            v = (v > 0.f) ? v : v * NEG_SLOPE;
            Eout[(size_t)(m0 + i + halfsel * 8) * EMBED_DIM + nb * 16 + nlane] = v;
        }
    }
}

// ------------------------------------------------------------- final gather
// Output: [user rows (4096 x 256)] then [item rows (4096 x 256)], each row is
// concat(emb, E1, E2, E3) of the selected node.
__global__ void ngcf_gather_kernel(const float* __restrict__ emb,
                                   const float* __restrict__ E1,
                                   const float* __restrict__ E2,
                                   const float* __restrict__ E3,
                                   const int* __restrict__ user_idx,
                                   const int* __restrict__ item_idx,
                                   float* __restrict__ out) {
    const int row = blockIdx.x;          // 0 .. 2*BATCH-1
    const int t   = threadIdx.x;         // 0 .. 255
    const int seg = t >> 6;              // which layer table
    const int dim = t & 63;

    int node;
    if (row < BATCH) node = user_idx[row];
    else             node = USER_NUM + item_idx[row - BATCH];

    const float* src = (seg == 0) ? emb : (seg == 1) ? E1 : (seg == 2) ? E2 : E3;
    out[(size_t)row * 256 + t] = src[(size_t)node * EMBED_DIM + dim];
}

// ------------------------------------------------------------------- launch
extern "C" void kernel_launch(void* const* d_in, const int* in_sizes, int n_in,
                              void* d_out, int out_size, void* d_ws, size_t ws_size,
                              hipStream_t stream) {
    const float* emb      = (const float*)d_in[0];   // NODE_NUM x 64
    const float* W1       = (const float*)d_in[1];   // 3 x 64 x 64
    const float* W2       = (const float*)d_in[2];   // 3 x 64 x 64
    const float* vals     = (const float*)d_in[3];   // NNZ
    const int*   row_idx  = (const int*)d_in[4];     // NNZ
    const int*   col_idx  = (const int*)d_in[5];     // NNZ
    const int*   user_idx = (const int*)d_in[6];     // BATCH
    const int*   item_idx = (const int*)d_in[7];     // BATCH

    const size_t EN = (size_t)NODE_NUM * EMBED_DIM;  // 6.4M floats = 25.6 MB
    float* side = (float*)d_ws;                      // needs 4*EN*4 = 102.4 MB
    float* E1   = side + EN;
    float* E2   = E1 + EN;
    float* E3   = E2 + EN;
    float* Eouts[N_LAYER] = {E1, E2, E3};

    const int layerBlocks = (NODE_TILES + 7) / 8;    // 8 waves / 256-thread block

    const float* Eprev = emb;
    for (int l = 0; l < N_LAYER; ++l) {
        ngcf_zero_kernel<<<(int)(EN / 4 / 256), 256, 0, stream>>>((float4*)side,
                                                                  (int)(EN / 4));
        ngcf_spmm_kernel<<<NNZ / 8, 256, 0, stream>>>(vals, row_idx, col_idx,
                                                      Eprev, side);
        ngcf_layer_kernel<<<layerBlocks, 256, 0, stream>>>(
            Eprev, side,
            W1 + (size_t)l * EMBED_DIM * EMBED_DIM,
            W2 + (size_t)l * EMBED_DIM * EMBED_DIM,
            Eouts[l]);
        Eprev = Eouts[l];
    }

    ngcf_gather_kernel<<<2 * BATCH, 256, 0, stream>>>(emb, E1, E2, E3,
                                                      user_idx, item_idx,
                                                      (float*)d_out);
}